// TemporalAttnDecoderRNN_20547123544565
// MI455X (gfx1250) — compile-verified
//
#include <hip/hip_runtime.h>
#include <hip/hip_bf16.h>
#include <math.h>

// Problem constants
#define B_  128
#define S_  400
#define T_  50
#define H_  512
#define E_  128
#define V_  50000
#define PAD_ 1000
#define VP_ (V_ + PAD_)   // 51000

typedef float v2f __attribute__((ext_vector_type(2)));
typedef float v8f __attribute__((ext_vector_type(8)));

__device__ __forceinline__ float wave_reduce_sum(float v) {
  #pragma unroll
  for (int off = 16; off > 0; off >>= 1) v += __shfl_xor(v, off, 32);
  return v;
}

// ---------------------------------------------------------------------------
// 1) GRU cell: x = emb[tok]; gx = x@W_ih.T + b_ih; gh = lh@W_hh.T + b_hh;
//    h = (1-z)*n + z*lh.   One block per batch row, 512 threads (one per h).
// ---------------------------------------------------------------------------
__global__ void __launch_bounds__(512)
gru_kernel(const int* __restrict__ tok, const float* __restrict__ emb,
           const float* __restrict__ last_h,
           const float* __restrict__ W_ih, const float* __restrict__ W_hh,
           const float* __restrict__ b_ih, const float* __restrict__ b_hh,
           float* __restrict__ h_ws) {
  const int b = blockIdx.x;
  const int i = threadIdx.x;              // 0..511
  __shared__ float x[E_];
  __shared__ float lh[H_];
  if (i < E_) x[i] = emb[(size_t)tok[b] * E_ + i];
  lh[i] = last_h[(size_t)b * H_ + i];
  __syncthreads();

  float gxr = b_ih[i], gxz = b_ih[H_ + i], gxn = b_ih[2 * H_ + i];
  const float* wr = W_ih + (size_t)i * E_;
  const float* wz = W_ih + (size_t)(H_ + i) * E_;
  const float* wn = W_ih + (size_t)(2 * H_ + i) * E_;
  for (int e = 0; e < E_; ++e) {
    float xe = x[e];
    gxr += xe * wr[e]; gxz += xe * wz[e]; gxn += xe * wn[e];
  }
  float ghr = b_hh[i], ghz = b_hh[H_ + i], ghn = b_hh[2 * H_ + i];
  const float* vr = W_hh + (size_t)i * H_;
  const float* vz = W_hh + (size_t)(H_ + i) * H_;
  const float* vn = W_hh + (size_t)(2 * H_ + i) * H_;
  for (int k = 0; k < H_; ++k) {
    float hk = lh[k];
    ghr += hk * vr[k]; ghz += hk * vz[k]; ghn += hk * vn[k];
  }
  float r = 1.f / (1.f + expf(-(gxr + ghr)));
  float z = 1.f / (1.f + expf(-(gxz + ghz)));
  float n = tanhf(gxn + r * ghn);
  h_ws[(size_t)b * H_ + i] = (1.f - z) * n + z * lh[i];
}

// ---------------------------------------------------------------------------
// 2) Fused attention: att scores, temporal normalization, decoder softmax,
//    encoder & decoder contexts. One block (256 threads = 8 waves) per batch.
// ---------------------------------------------------------------------------
__global__ void __launch_bounds__(256)
attention_kernel(const float* __restrict__ h_ws, const float* __restrict__ enc,
                 const float* __restrict__ prev_att, const float* __restrict__ prev_dec,
                 const float* __restrict__ w_h, const float* __restrict__ w_d,
                 float* __restrict__ att_scores_ws, float* __restrict__ att_dist_out,
                 float* __restrict__ enc_ctx, float* __restrict__ dec_ctx) {
  const int b = blockIdx.x;
  const int tid = threadIdx.x;
  const int wave = tid >> 5, lane = tid & 31;
  __shared__ float h_sh[H_];
  __shared__ float sc[S_];
  __shared__ float adist[S_];
  __shared__ float dsc[T_ + 14];
  __shared__ float red[8];

  for (int i = tid; i < H_; i += 256) h_sh[i] = h_ws[(size_t)b * H_ + i];
  __syncthreads();

  // att_scores[b,s] = sum_h h*w_h*enc  (wave per s)
  for (int s = wave; s < S_; s += 8) {
    const float* ep = enc + ((size_t)b * S_ + s) * H_;
    float acc = 0.f;
    for (int hh = lane; hh < H_; hh += 32) acc += h_sh[hh] * w_h[hh] * ep[hh];
    acc = wave_reduce_sum(acc);
    if (lane == 0) { sc[s] = acc; att_scores_ws[(size_t)b * S_ + s] = acc; }
  }
  __syncthreads();

  // temporal = exp(sc)/sum_t exp(prev_att);  then normalize over s
  float part = 0.f;
  for (int s = tid; s < S_; s += 256) {
    const float* pp = prev_att + (size_t)b * T_ * S_ + s;
    float denom = 0.f;
    for (int t = 0; t < T_; ++t) denom += expf(pp[(size_t)t * S_]);
    float tmp = expf(sc[s]) / denom;
    adist[s] = tmp;
    part += tmp;
  }
  part = wave_reduce_sum(part);
  if (lane == 0) red[wave] = part;
  __syncthreads();
  float total = 0.f;
  #pragma unroll
  for (int i = 0; i < 8; ++i) total += red[i];
  const float inv = 1.f / total;
  for (int s = tid; s < S_; s += 256) {
    float a = adist[s] * inv;
    adist[s] = a;
    att_dist_out[(size_t)b * S_ + s] = a;
  }
  __syncthreads();

  // decoder scores (wave per t) then softmax over 50
  for (int t = wave; t < T_; t += 8) {
    const float* dp = prev_dec + ((size_t)b * T_ + t) * H_;
    float acc = 0.f;
    for (int hh = lane; hh < H_; hh += 32) acc += h_sh[hh] * w_d[hh] * dp[hh];
    acc = wave_reduce_sum(acc);
    if (lane == 0) dsc[t] = acc;
  }
  __syncthreads();
  float m = -INFINITY;
  for (int t = 0; t < T_; ++t) m = fmaxf(m, dsc[t]);
  float ssum = 0.f;
  for (int t = 0; t < T_; ++t) ssum += expf(dsc[t] - m);
  __syncthreads();
  if (tid < T_) dsc[tid] = expf(dsc[tid] - m) / ssum;
  __syncthreads();

  // contexts (coalesced over hh)
  for (int hh = tid; hh < H_; hh += 256) {
    float ev = 0.f;
    for (int s = 0; s < S_; ++s) ev += adist[s] * enc[((size_t)b * S_ + s) * H_ + hh];
    enc_ctx[(size_t)b * H_ + hh] = ev;
    float dv = 0.f;
    for (int t = 0; t < T_; ++t) dv += dsc[t] * prev_dec[((size_t)b * T_ + t) * H_ + hh];
    dec_ctx[(size_t)b * H_ + hh] = dv;
  }
}

// ---------------------------------------------------------------------------
// 3) combined = [h, enc_ctx, dec_ctx, ctrl];  hidden = combined@outh_W.T + b;
//    p_gen = sigmoid(combined@gen_W.T + gen_b).  Wave-per-output reductions.
// ---------------------------------------------------------------------------
__global__ void __launch_bounds__(256)
combined_kernel(const float* __restrict__ h_ws, const float* __restrict__ enc_ctx,
                const float* __restrict__ dec_ctx, const float* __restrict__ ctrl,
                const float* __restrict__ outh_W, const float* __restrict__ outh_b,
                const float* __restrict__ gen_W, const float* __restrict__ gen_b,
                float* __restrict__ hidden_ws, float* __restrict__ pgen_out) {
  const int b = blockIdx.x;
  const int tid = threadIdx.x;
  const int wave = tid >> 5, lane = tid & 31;
  __shared__ float comb[3 * H_ + 1];
  for (int i = tid; i < H_; i += 256) {
    comb[i]          = h_ws[(size_t)b * H_ + i];
    comb[H_ + i]     = enc_ctx[(size_t)b * H_ + i];
    comb[2 * H_ + i] = dec_ctx[(size_t)b * H_ + i];
  }
  if (tid == 0) comb[3 * H_] = ctrl[b];
  __syncthreads();
  for (int j = wave; j < E_ + 1; j += 8) {
    const float* wrow = (j < E_) ? (outh_W + (size_t)j * (3 * H_ + 1)) : gen_W;
    float acc = 0.f;
    for (int c = lane; c < 3 * H_ + 1; c += 32) acc += comb[c] * wrow[c];
    acc = wave_reduce_sum(acc);
    if (lane == 0) {
      if (j < E_) hidden_ws[(size_t)b * E_ + j] = acc + outh_b[j];
      else        pgen_out[b] = 1.f / (1.f + expf(-(acc + gen_b[0])));
    }
  }
}

// ---------------------------------------------------------------------------
// 4) Vocab GEMM with V_WMMA_F32_16X16X4_F32:
//    logits(128x50000) = hidden(128x128) @ outv_W.T + outv_b.
//    One wave per 16x16 output tile; 32 WMMA issues (K=128, 4/issue).
//    outv_W (25.6 MB) is L2-resident across the 8 M-tiles (192 MB L2).
//    Fragment layouts per ISA 7.12.2 (32-bit A 16x4 / mirrored B / v8f C-D).
// ---------------------------------------------------------------------------
__global__ void __launch_bounds__(32)
vocab_gemm_wmma(const float* __restrict__ hidden, const float* __restrict__ outv_W,
                const float* __restrict__ outv_b, float* __restrict__ logits) {
  const int n0 = blockIdx.x * 16;          // 3125 tiles over V
  const int m0 = blockIdx.y * 16;          // 8 tiles over B
  const int lane = threadIdx.x;
  const int half = lane >> 4;              // selects K pair {0,1} vs {2,3}
  const int l15  = lane & 15;

  const float* arow = hidden + (size_t)(m0 + l15) * E_ + 2 * half;  // lanes hold M
  const float* brow = outv_W + (size_t)(n0 + l15) * E_ + 2 * half;  // lanes hold N (B[k][n]=outv_W[n][k])

  v8f acc = {};
  #pragma unroll 4
  for (int k0 = 0; k0 < E_; k0 += 4) {
    v2f a, bf;
    a.x  = arow[k0]; a.y  = arow[k0 + 1];
    bf.x = brow[k0]; bf.y = brow[k0 + 1];
    acc = __builtin_amdgcn_wmma_f32_16x16x4_f32(
        /*neg_a=*/false, a, /*neg_b=*/false, bf,
        /*c_mod=*/(short)0, acc, /*reuse_a=*/false, /*reuse_b=*/false);
  }
  const int col = n0 + l15;
  const float bias = outv_b[col];
  #pragma unroll
  for (int i = 0; i < 8; ++i) {
    const int row = m0 + i + half * 8;     // C/D: vgpr i -> M=i (lanes<16) / M=i+8
    logits[(size_t)row * V_ + col] = acc[i] + bias;
  }
}

// ---------------------------------------------------------------------------
// 5) Per-row softmax stats over V (max + 1/sum(exp)).
// ---------------------------------------------------------------------------
__global__ void __launch_bounds__(256)
row_softmax_stats(const float* __restrict__ logits, float* __restrict__ rowmax,
                  float* __restrict__ rowinv) {
  const int b = blockIdx.x;
  const int tid = threadIdx.x;
  __shared__ float red[256];
  const float* lp = logits + (size_t)b * V_;
  float m = -INFINITY;
  for (int j = tid; j < V_; j += 256) m = fmaxf(m, lp[j]);
  red[tid] = m; __syncthreads();
  for (int s = 128; s > 0; s >>= 1) {
    if (tid < s) red[tid] = fmaxf(red[tid], red[tid + s]);
    __syncthreads();
  }
  m = red[0]; __syncthreads();
  float sum = 0.f;
  for (int j = tid; j < V_; j += 256) sum += expf(lp[j] - m);
  red[tid] = sum; __syncthreads();
  for (int s = 128; s > 0; s >>= 1) {
    if (tid < s) red[tid] += red[tid + s];
    __syncthreads();
  }
  if (tid == 0) { rowmax[b] = m; rowinv[b] = 1.f / red[0]; }
}

// ---------------------------------------------------------------------------
// 6) p_vocab = softmax(logits); p_final base = pad(p_vocab*p_gen).
// ---------------------------------------------------------------------------
__global__ void __launch_bounds__(256)
finalize_kernel(const float* __restrict__ logits, const float* __restrict__ rowmax,
                const float* __restrict__ rowinv, const float* __restrict__ pgen,
                float* __restrict__ pvocab_out, float* __restrict__ pfinal_out) {
  const size_t idx = (size_t)blockIdx.x * 256 + threadIdx.x;
  const size_t total = (size_t)B_ * VP_;
  if (idx >= total) return;
  const int b = (int)(idx / VP_);
  const int c = (int)(idx % VP_);
  if (c < V_) {
    float pv = expf(logits[(size_t)b * V_ + c] - rowmax[b]) * rowinv[b];
    pvocab_out[(size_t)b * V_ + c] = pv;
    pfinal_out[idx] = pv * pgen[b];
  } else {
    pfinal_out[idx] = 0.f;
  }
}

// ---------------------------------------------------------------------------
// 7) Pointer-generator scatter: p_final[b, fiv[b,s]] += (1-p_gen)*att_dist.
// ---------------------------------------------------------------------------
__global__ void __launch_bounds__(256)
scatter_kernel(const int* __restrict__ full_input, const float* __restrict__ att_dist,
               const float* __restrict__ pgen, float* __restrict__ pfinal) {
  const int idx = blockIdx.x * 256 + threadIdx.x;
  if (idx >= B_ * S_) return;
  const int b = idx / S_;
  const int v = full_input[idx];
  atomicAdd(&pfinal[(size_t)b * VP_ + v], (1.f - pgen[b]) * att_dist[idx]);
}

// ---------------------------------------------------------------------------
// 8) Pack concatenated outputs: decoder_h_states, new_previous_att, h.
// ---------------------------------------------------------------------------
__global__ void __launch_bounds__(256)
pack_kernel(const float* __restrict__ prev_dec, const float* __restrict__ prev_att,
            const float* __restrict__ h_ws, const float* __restrict__ scores_ws,
            float* __restrict__ dec_out, float* __restrict__ att_out,
            float* __restrict__ h_out) {
  const size_t n1 = (size_t)B_ * (T_ + 1) * H_;
  const size_t n2 = (size_t)B_ * (T_ + 1) * S_;
  const size_t n3 = (size_t)B_ * H_;
  size_t idx = (size_t)blockIdx.x * 256 + threadIdx.x;
  if (idx < n1) {
    size_t b = idx / ((T_ + 1) * H_), r = idx % ((T_ + 1) * H_);
    size_t t = r / H_, hh = r % H_;
    dec_out[idx] = (t < T_) ? prev_dec[(b * T_ + t) * H_ + hh] : h_ws[b * H_ + hh];
  } else if (idx < n1 + n2) {
    size_t j = idx - n1;
    size_t b = j / ((T_ + 1) * S_), r = j % ((T_ + 1) * S_);
    size_t t = r / S_, s = r % S_;
    att_out[j] = (t < T_) ? prev_att[(b * T_ + t) * S_ + s] : scores_ws[b * S_ + s];
  } else if (idx < n1 + n2 + n3) {
    size_t j = idx - n1 - n2;
    h_out[j] = h_ws[j];
  }
}

// ---------------------------------------------------------------------------
extern "C" void kernel_launch(void* const* d_in, const int* in_sizes, int n_in,
                              void* d_out, int out_size, void* d_ws, size_t ws_size,
                              hipStream_t stream) {
  // Inputs, in setup_inputs() order
  const int*   input_token = (const int*)  d_in[0];
  const float* prev_dec    = (const float*)d_in[1];
  const float* last_h      = (const float*)d_in[2];
  const float* enc         = (const float*)d_in[3];
  const int*   full_input  = (const int*)  d_in[4];
  const float* prev_att    = (const float*)d_in[5];
  const float* ctrl        = (const float*)d_in[6];
  const float* emb         = (const float*)d_in[7];
  const float* W_ih        = (const float*)d_in[8];
  const float* W_hh        = (const float*)d_in[9];
  const float* b_ih        = (const float*)d_in[10];
  const float* b_hh        = (const float*)d_in[11];
  const float* w_h         = (const float*)d_in[12];
  const float* w_d         = (const float*)d_in[13];
  const float* gen_W       = (const float*)d_in[14];
  const float* gen_b       = (const float*)d_in[15];
  const float* outh_W      = (const float*)d_in[16];
  const float* outh_b      = (const float*)d_in[17];
  const float* outv_W      = (const float*)d_in[18];
  const float* outv_b      = (const float*)d_in[19];

  // Output layout (flat concat in reference return order)
  float* out = (float*)d_out;
  const size_t o_pfinal   = 0;
  const size_t o_pgen     = o_pfinal + (size_t)B_ * VP_;
  const size_t o_pvocab   = o_pgen + B_;
  const size_t o_attdist  = o_pvocab + (size_t)B_ * V_;
  const size_t o_decstate = o_attdist + (size_t)B_ * S_;
  const size_t o_h        = o_decstate + (size_t)B_ * (T_ + 1) * H_;
  const size_t o_prevatt  = o_h + (size_t)B_ * H_;

  // Workspace layout (floats); logits dominate (~25.6 MB)
  float* ws = (float*)d_ws;
  float* ws_h      = ws;                                  // B*H
  float* ws_scores = ws_h + (size_t)B_ * H_;              // B*S
  float* ws_encctx = ws_scores + (size_t)B_ * S_;         // B*H
  float* ws_decctx = ws_encctx + (size_t)B_ * H_;         // B*H
  float* ws_hidden = ws_decctx + (size_t)B_ * H_;         // B*E
  float* ws_rowmax = ws_hidden + (size_t)B_ * E_;         // B
  float* ws_rowinv = ws_rowmax + B_;                      // B
  float* ws_logits = ws_rowinv + B_;                      // B*V

  // 1) GRU cell
  gru_kernel<<<dim3(B_), dim3(512), 0, stream>>>(
      input_token, emb, last_h, W_ih, W_hh, b_ih, b_hh, ws_h);

  // 2) Fused attention + contexts (writes att_dist directly to d_out)
  attention_kernel<<<dim3(B_), dim3(256), 0, stream>>>(
      ws_h, enc, prev_att, prev_dec, w_h, w_d,
      ws_scores, out + o_attdist, ws_encctx, ws_decctx);

  // 3) Output projection + p_gen (p_gen written directly to d_out)
  combined_kernel<<<dim3(B_), dim3(256), 0, stream>>>(
      ws_h, ws_encctx, ws_decctx, ctrl, outh_W, outh_b, gen_W, gen_b,
      ws_hidden, out + o_pgen);

  // 4) WMMA vocab GEMM -> logits
  vocab_gemm_wmma<<<dim3(V_ / 16, B_ / 16), dim3(32), 0, stream>>>(
      ws_hidden, outv_W, outv_b, ws_logits);

  // 5) Softmax stats per row
  row_softmax_stats<<<dim3(B_), dim3(256), 0, stream>>>(
      ws_logits, ws_rowmax, ws_rowinv);

  // 6) p_vocab + p_final base
  {
    const size_t total = (size_t)B_ * VP_;
    finalize_kernel<<<dim3((unsigned)((total + 255) / 256)), dim3(256), 0, stream>>>(
        ws_logits, ws_rowmax, ws_rowinv, out + o_pgen,
        out + o_pvocab, out + o_pfinal);
  }

  // 7) Pointer-generator scatter-add
  scatter_kernel<<<dim3((B_ * S_ + 255) / 256), dim3(256), 0, stream>>>(
      full_input, out + o_attdist, out + o_pgen, out + o_pfinal);

  // 8) Pack concatenated state outputs
  {
    const size_t total = (size_t)B_ * (T_ + 1) * H_ + (size_t)B_ * (T_ + 1) * S_ +
                         (size_t)B_ * H_;
    pack_kernel<<<dim3((unsigned)((total + 255) / 256)), dim3(256), 0, stream>>>(
        prev_dec, prev_att, ws_h, ws_scores,
        out + o_decstate, out + o_prevatt, out + o_h);
  }
}